// LinearRNN_19456201851458
// MI455X (gfx1250) — compile-verified
//
#include <hip/hip_runtime.h>

typedef __attribute__((ext_vector_type(2))) float    v2f;
typedef __attribute__((ext_vector_type(8))) float    v8f;
typedef __attribute__((ext_vector_type(4))) unsigned v4u;
typedef __attribute__((ext_vector_type(4))) int      v4i;
typedef __attribute__((ext_vector_type(8))) int      v8i;

#define HDIM 1024
#define BDIM 32
#define LDIM 512
#define PRED 32
#define LN_EPS 1e-5f

#define RNN_NWG 16          // persistent WGs; 64 output columns each; 1 WG/WGP (LDS-bound)
#define RNN_THREADS 128     // 4 waves; each wave: 1 row-tile x 2 col-tiles (shared A frag)
#define ROWS_PER_WG 64      // Whh rows resident in LDS per WG
#define BS_ROW 1028         // LDS row pitch in dwords: 1024 + 4 pad dwords (TDM pad layout)

#if __has_builtin(__builtin_amdgcn_tensor_load_to_lds)
#define USE_TDM 1
#endif

#define WMMA_F32(a, b, c) \
  __builtin_amdgcn_wmma_f32_16x16x4_f32(false, (a), false, (b), (short)0, (c), false, false)

// --------------------------------------------------------------------------
// Kernel 1: xW[l][b][:] = x[b][l][:] @ Wxh^T   (M = L*B = 16384, N = K = 1024)
// 32x32 register tile per wave: 4 accumulators, A/B fragments reused 2x each
// (8 FLOP per L2 byte instead of 4).
// --------------------------------------------------------------------------
__global__ __launch_bounds__(256) void xw_gemm_kernel(const float* __restrict__ x,
                                                      const float* __restrict__ Wxh,
                                                      float* __restrict__ xW) {
  const int wave = threadIdx.x >> 5;
  const int lane = threadIdx.x & 31;
  const int tile = blockIdx.x * 8 + wave;     // 16384 tiles: 512 l-groups x 32 col-groups
  const int cg   = tile & 31;                 // 32-column group
  const int l    = tile >> 5;                 // sequence position
  const int li   = lane & 15;
  const int hk   = (lane >> 4) * 2;           // K sub-offset inside group of 4

  const float* a0 = x + ((size_t)li * LDIM + l) * HDIM;          // A rows b = li
  const float* a1 = x + ((size_t)(16 + li) * LDIM + l) * HDIM;   // A rows b = 16+li
  const float* b0 = Wxh + (size_t)(cg * 32 + li) * HDIM;         // B col n = Wxh row n
  const float* b1 = Wxh + (size_t)(cg * 32 + 16 + li) * HDIM;

  v8f c00 = {}, c01 = {}, c10 = {}, c11 = {};
  #pragma unroll 2
  for (int k0 = 0; k0 < HDIM; k0 += 4) {
    v2f A0 = *(const v2f*)(a0 + k0 + hk);
    v2f A1 = *(const v2f*)(a1 + k0 + hk);
    v2f B0 = *(const v2f*)(b0 + k0 + hk);
    v2f B1 = *(const v2f*)(b1 + k0 + hk);
    c00 = WMMA_F32(A0, B0, c00);
    c01 = WMMA_F32(A0, B1, c01);
    c10 = WMMA_F32(A1, B0, c10);
    c11 = WMMA_F32(A1, B1, c11);
  }
  const int mrow = (lane >> 4) * 8;  // D: VGPR r -> M = r + (lane>=16 ? 8 : 0), N = li
  float* o00 = xW + ((size_t)l * BDIM + mrow) * HDIM + cg * 32 + li;
  float* o10 = xW + ((size_t)l * BDIM + 16 + mrow) * HDIM + cg * 32 + li;
  #pragma unroll
  for (int r = 0; r < 8; ++r) {
    o00[(size_t)r * HDIM]      = c00[r];
    o00[(size_t)r * HDIM + 16] = c01[r];
    o10[(size_t)r * HDIM]      = c10[r];
    o10[(size_t)r * HDIM + 16] = c11[r];
  }
}

// --------------------------------------------------------------------------
// Grid-wide barrier (persistent kernel, all WGs co-resident: 1 per WGP).
// Acquire atomic load invalidates WGP$ so fresh h_{t-1} is visible.
// --------------------------------------------------------------------------
__device__ inline void grid_barrier(unsigned* cnt, unsigned* gen,
                                    unsigned target, unsigned nwg) {
  __syncthreads();
  if (threadIdx.x == 0) {
    __threadfence();  // make this WG's h_t stores visible device-wide
    unsigned arrived = __hip_atomic_fetch_add(cnt, 1u, __ATOMIC_ACQ_REL,
                                              __HIP_MEMORY_SCOPE_AGENT);
    if (arrived == nwg - 1u) {
      __hip_atomic_store(cnt, 0u, __ATOMIC_RELAXED, __HIP_MEMORY_SCOPE_AGENT);
      __hip_atomic_fetch_add(gen, 1u, __ATOMIC_RELEASE, __HIP_MEMORY_SCOPE_AGENT);
    } else {
      while (__hip_atomic_load(gen, __ATOMIC_ACQUIRE,
                               __HIP_MEMORY_SCOPE_AGENT) < target)
        __builtin_amdgcn_s_sleep(1);
    }
  }
  __syncthreads();
}

// --------------------------------------------------------------------------
// Kernel 2: persistent linear recurrence.
//   t in [0,512):  rec[:,t,:]      = xW[t] + h_{t-1} @ Whh^T
//   t in [512,544): predRaw[:,j,:] =          h_{t-1} @ Whh^T
// Each WG pins 64 Whh rows (257 KB) in LDS for all 544 steps, staged by the
// Tensor Data Mover with its pad feature building the conflict-free pitch:
// LDS dword offset of (row r, col k) = r*1028 + k + k/256.
// --------------------------------------------------------------------------
__global__ __launch_bounds__(RNN_THREADS) void rnn_kernel(
    const float* __restrict__ xW, const float* __restrict__ Whh,
    float* __restrict__ rec,        // d_out, [B][L][H]
    float* __restrict__ predRaw,    // ws,    [B][PRED][H]
    unsigned* __restrict__ sync) {  // sync[0]=counter sync[1]=generation
  __shared__ float Bs[ROWS_PER_WG * BS_ROW];  // 257 KB of the 320 KB WGP LDS

  const int wg   = blockIdx.x;
  const int wave = threadIdx.x >> 5;
  const int lane = threadIdx.x & 31;
  const int li   = lane & 15;
  const int hk   = (lane >> 4) * 2;

#ifdef USE_TDM
  if (wave == 0) {
    // TDM descriptor: 2D tile, 64 rows x 1024 cols of 4B data, LDS padding of
    // 1 dword after every 256 dwords -> row pitch 1028 dwords.
    const unsigned ldsBase = (unsigned)(unsigned long long)(&Bs[0]);  // low 32b = LDS offset
    const unsigned long long ga =
        (unsigned long long)(const void*)(Whh + (size_t)wg * ROWS_PER_WG * HDIM);
    v4u g0 = { 1u,                                         // count=1 (valid), not restore
               ldsBase,                                    // lds_addr (bytes)
               (unsigned)(ga & 0xFFFFFFFFu),               // global_addr[31:0]
               (unsigned)((ga >> 32) & 0x1FFFFFFu) | (2u << 30) };  // addr[56:32] | type=2
    v8i g1 = { (int)((2u << 16) | (1u << 20) | (7u << 22)),// data_size=4B, pad_en, ivl=256dw, amt=1dw
               (int)(1024u << 16),                         // tensor_dim0 = 1024
               (int)(ROWS_PER_WG << 16),                   // tensor_dim1 = 64
               (int)(1024u << 16),                         // tile_dim0 = 1024
               (int)ROWS_PER_WG,                           // tile_dim1 = 64, tile_dim2 = 0
               1024, 0, 0 };                               // tensor_dim0_stride = 1024
    v4i gz = { 0, 0, 0, 0 };
#if defined(__clang_major__) && (__clang_major__ >= 23)
    v8i gz8 = { 0, 0, 0, 0, 0, 0, 0, 0 };
    __builtin_amdgcn_tensor_load_to_lds(g0, g1, gz, gz, gz8, 0);
#else
    __builtin_amdgcn_tensor_load_to_lds(g0, g1, gz, gz, 0);
#endif
    __builtin_amdgcn_s_wait_tensorcnt(0);
  }
#else
  for (int i = threadIdx.x; i < ROWS_PER_WG * HDIM; i += RNN_THREADS) {
    const int r = i >> 10, c = i & 1023;
    Bs[r * BS_ROW + c + (c >> 8)] = Whh[(size_t)(wg * ROWS_PER_WG + r) * HDIM + c];
  }
#endif
  __syncthreads();

  const int m0 = (wave & 1) * 16;            // row-tile base (batch index)
  const int cp = wave >> 1;                  // column pair 0/1 (32 cols each)
  const int n0 = cp * 32;                    // LDS row base of acc0; acc1 at +16
  const int nG0 = wg * 64 + n0 + li;         // global column of acc0 lane slot
  const float* b0p = &Bs[(n0 + li) * BS_ROW + hk];
  const float* b1p = &Bs[(n0 + 16 + li) * BS_ROW + hk];
  const int mrow = (lane >> 4) * 8;

  for (int t = 0; t < LDIM + PRED; ++t) {
    v8f acc0 = {}, acc1 = {};
    if (t > 0) {
      const float* hbase;
      size_t rowStride;
      if (t <= LDIM) { hbase = rec + (size_t)(t - 1) * HDIM;            rowStride = (size_t)LDIM * HDIM; }
      else           { hbase = predRaw + (size_t)(t - 1 - LDIM) * HDIM; rowStride = (size_t)PRED * HDIM; }
      const float* afrag = hbase + (size_t)(m0 + li) * rowStride + hk;
      #pragma unroll
      for (int kb = 0; kb < 4; ++kb) {       // K blocked by 256 so pad offset is invariant
        const float* ab  = afrag + kb * 256;
        const float* bb0 = b0p + kb * 257;
        const float* bb1 = b1p + kb * 257;
        #pragma unroll 4
        for (int kk = 0; kk < 256; kk += 4) {
          v2f a  = *(const v2f*)(ab + kk);    // h_{t-1} from global (L2)
          v2f B0 = *(const v2f*)(bb0 + kk);   // Whh^T from LDS (conflict-free)
          v2f B1 = *(const v2f*)(bb1 + kk);
          acc0 = WMMA_F32(a, B0, acc0);
          acc1 = WMMA_F32(a, B1, acc1);
        }
      }
    }
    if (t < LDIM) {
      const float* xwp = xW + ((size_t)t * BDIM + m0 + mrow) * HDIM + nG0;
      float* outp = rec + (size_t)(m0 + mrow) * LDIM * HDIM + (size_t)t * HDIM + nG0;
      #pragma unroll
      for (int r = 0; r < 8; ++r) {
        outp[(size_t)r * LDIM * HDIM]      = acc0[r] + xwp[(size_t)r * HDIM];
        outp[(size_t)r * LDIM * HDIM + 16] = acc1[r] + xwp[(size_t)r * HDIM + 16];
      }
    } else {
      const int j = t - LDIM;
      float* outp = predRaw + (size_t)(m0 + mrow) * PRED * HDIM + (size_t)j * HDIM + nG0;
      #pragma unroll
      for (int r = 0; r < 8; ++r) {
        outp[(size_t)r * PRED * HDIM]      = acc0[r];
        outp[(size_t)r * PRED * HDIM + 16] = acc1[r];
      }
    }
    grid_barrier(&sync[0], &sync[1], (unsigned)(t + 1), (unsigned)gridDim.x);
  }
}

// --------------------------------------------------------------------------
// Kernel 3: LayerNorm over last dim of predRaw -> d_out second region.
// --------------------------------------------------------------------------
__global__ __launch_bounds__(256) void ln_kernel(const float* __restrict__ yIn,
                                                 const float* __restrict__ lnw,
                                                 const float* __restrict__ lnb,
                                                 float* __restrict__ out) {
  __shared__ float sh_s[8], sh_s2[8];
  const int row = blockIdx.x;
  const float* y = yIn + (size_t)row * HDIM;
  float4 v = *(const float4*)(y + threadIdx.x * 4);
  float s  = v.x + v.y + v.z + v.w;
  float s2 = v.x * v.x + v.y * v.y + v.z * v.z + v.w * v.w;
  #pragma unroll
  for (int off = 16; off > 0; off >>= 1) {
    s  += __shfl_down(s, off, 32);
    s2 += __shfl_down(s2, off, 32);
  }
  if ((threadIdx.x & 31) == 0) { sh_s[threadIdx.x >> 5] = s; sh_s2[threadIdx.x >> 5] = s2; }
  __syncthreads();
  if (threadIdx.x == 0) {
    float a = 0.f, b = 0.f;
    #pragma unroll
    for (int i = 0; i < 8; ++i) { a += sh_s[i]; b += sh_s2[i]; }
    sh_s[0] = a; sh_s2[0] = b;
  }
  __syncthreads();
  const float mean = sh_s[0] * (1.0f / HDIM);
  const float var  = sh_s2[0] * (1.0f / HDIM) - mean * mean;
  const float inv  = rsqrtf(var + LN_EPS);
  const int k = threadIdx.x * 4;
  float4 w4 = *(const float4*)(lnw + k);
  float4 b4 = *(const float4*)(lnb + k);
  float4 o;
  o.x = (v.x - mean) * inv * w4.x + b4.x;
  o.y = (v.y - mean) * inv * w4.y + b4.y;
  o.z = (v.z - mean) * inv * w4.z + b4.z;
  o.w = (v.w - mean) * inv * w4.w + b4.w;
  *(float4*)(out + (size_t)row * HDIM + k) = o;
}

__global__ void init_sync_kernel(unsigned* sync) { sync[0] = 0u; sync[1] = 0u; }

// --------------------------------------------------------------------------
extern "C" void kernel_launch(void* const* d_in, const int* in_sizes, int n_in,
                              void* d_out, int out_size, void* d_ws, size_t ws_size,
                              hipStream_t stream) {
  const float* x    = (const float*)d_in[0];
  const float* Wxh  = (const float*)d_in[1];
  const float* Whh  = (const float*)d_in[2];
  const float* ln_w = (const float*)d_in[3];
  const float* ln_b = (const float*)d_in[4];
  // d_in[5] = pred_len (==32, compile-time constant here)

  float* rec  = (float*)d_out;                                    // [B][L][H]
  float* outs = (float*)d_out + (size_t)BDIM * LDIM * HDIM;       // [B][PRED][H]

  float* xW       = (float*)d_ws;                                 // [L][B][H] 64 MB
  float* predRaw  = xW + (size_t)LDIM * BDIM * HDIM;              // [B][PRED][H] 4 MB
  unsigned* syncp = (unsigned*)(predRaw + (size_t)BDIM * PRED * HDIM);

  init_sync_kernel<<<1, 1, 0, stream>>>(syncp);
  xw_gemm_kernel<<<2048, 256, 0, stream>>>(x, Wxh, xW);
  rnn_kernel<<<RNN_NWG, RNN_THREADS, 0, stream>>>(xW, Whh, rec, predRaw, syncp);
  ln_kernel<<<BDIM * PRED, 256, 0, stream>>>(predRaw, ln_w, ln_b, outs);
}